// fclayer_25563645346508
// MI455X (gfx1250) — compile-verified
//
#include <hip/hip_runtime.h>

// CDNA5 / gfx1250, wave32. Batched per-channel dot product:
//   out[b, n] = dot(x[b, n, 0:128], W[n, 0:128]) + bias[n]
// Memory-bound (AI ~0.5 flop/byte): stream x once at HBM rate, compute via
// V_WMMA_F32_16X16X4_F32 with masked B columns (one channel per column slot).

typedef __attribute__((ext_vector_type(2))) float v2f;
typedef __attribute__((ext_vector_type(8))) float v8f;

#define FC_OUT_N 2048
#define FC_D     128
#define WAVES_PER_BLOCK 8

__global__ __launch_bounds__(32 * WAVES_PER_BLOCK)
void fclayer_wmma_kernel(const float* __restrict__ x,
                         const float* __restrict__ W,
                         const float* __restrict__ bias,
                         float* __restrict__ out,
                         int B)
{
    const int lane  = threadIdx.x & 31;
    const int lrow  = lane & 15;   // A: batch row M (lanes 0-15 / 16-31); B,C: column N
    const int lhalf = lane >> 4;   // 0 or 1: K sub-offset selector for A/B fragments

    const int wave   = blockIdx.x * WAVES_PER_BLOCK + (threadIdx.x >> 5);
    const int nTiles = FC_OUT_N / 16;            // 128
    const int n0 = (wave % nTiles) * 16;
    const int b0 = (wave / nTiles) * 16;
    if (b0 + 16 > B) return;                     // uniform per wave; EXEC stays all-1s

    // Two accumulators (even/odd channel slot) to shorten the WMMA RAW chain.
    v8f acc0 = {0.f, 0.f, 0.f, 0.f, 0.f, 0.f, 0.f, 0.f};
    v8f acc1 = {0.f, 0.f, 0.f, 0.f, 0.f, 0.f, 0.f, 0.f};

    // Per-lane base: row x[b0+lrow, n, :]; lanes 16-31 shifted by 2 floats so a
    // single b64 load per K-step yields the 16x4 f32 A fragment layout:
    //   VGPR0: K=k0 (lanes 0-15) / K=k0+2 (lanes 16-31); VGPR1: K=k0+1 / k0+3.
    const size_t rowBase = ((size_t)(b0 + lrow) * FC_OUT_N) * FC_D + 2 * lhalf;

    for (int j = 0; j < 16; ++j) {
        const int n = n0 + j;
        const float* xr = x + rowBase + (size_t)n * FC_D;
        const float* wr = W + (size_t)n * FC_D + 2 * lhalf;
        const bool sel = (lrow == j);            // B column mask (data mask, not EXEC)

        // Prefetch the next channel's x row (speculative; dropped if OOB).
        __builtin_prefetch(xr + FC_D, 0, 0);

        v8f c = (j & 1) ? acc1 : acc0;
        #pragma unroll
        for (int k0 = 0; k0 < FC_D; k0 += 4) {
            v2f a  = *(const v2f*)(xr + k0);     // global_load_b64, 8B-aligned
            v2f wv = *(const v2f*)(wr + k0);     // W row: L2-hot (1 MB total)
            v2f bm;
            bm.x = sel ? wv.x : 0.0f;            // B[k0+2*lhalf  , lrow]
            bm.y = sel ? wv.y : 0.0f;            // B[k0+2*lhalf+1, lrow]
            // D = A(16x4) * B(4x16) + C : only column j receives nonzero updates.
            c = __builtin_amdgcn_wmma_f32_16x16x4_f32(
                    /*neg_a=*/false, a, /*neg_b=*/false, bm,
                    /*c_mod=*/(short)0, c, /*reuse_a=*/false, /*reuse_b=*/false);
        }
        if (j & 1) acc1 = c; else acc0 = c;
    }

    // Both accumulators hold disjoint (zero-elsewhere) column sets: sum them.
    // C/D layout: VGPR v, lane l -> element (M = v + 8*(l>>4), N = l&15).
    const float bv = bias[n0 + lrow];
    #pragma unroll
    for (int v = 0; v < 8; ++v) {
        float r = acc0[v] + acc1[v] + bv;
        out[(size_t)(b0 + v + 8 * lhalf) * FC_OUT_N + (n0 + lrow)] = r;
    }
}

extern "C" void kernel_launch(void* const* d_in, const int* in_sizes, int n_in,
                              void* d_out, int out_size, void* d_ws, size_t ws_size,
                              hipStream_t stream) {
    (void)n_in; (void)out_size; (void)d_ws; (void)ws_size;
    const float* x    = (const float*)d_in[0];
    const float* W    = (const float*)d_in[1];
    const float* bias = (const float*)d_in[2];
    float*       out  = (float*)d_out;

    const int B = in_sizes[0] / (FC_OUT_N * FC_D);     // 256
    const int waves  = (B / 16) * (FC_OUT_N / 16);     // 2048 output tiles, 1 wave each
    const int blocks = (waves + WAVES_PER_BLOCK - 1) / WAVES_PER_BLOCK;  // 256

    fclayer_wmma_kernel<<<blocks, 32 * WAVES_PER_BLOCK, 0, stream>>>(x, W, bias, out, B);
}